// DGCNN_26036091748787
// MI455X (gfx1250) — compile-verified
//
#include <hip/hip_runtime.h>
#include <hip/hip_bf16.h>
#include <math.h>

typedef __attribute__((ext_vector_type(2))) float v2f;
typedef __attribute__((ext_vector_type(8))) float v8f;
typedef __attribute__((ext_vector_type(4))) unsigned int u32x4;
typedef __attribute__((ext_vector_type(8))) int i32x8;
typedef __attribute__((ext_vector_type(4))) int i32x4;

#define N_NODES 131072
#define NPG     512
#define B_GR    256
#define E_EDGES 2097152
#define EF      8
#define EMB     256
#define HID     32
#define TOT     97
#define K_POOL  30
#define S_SEQ   5
#define V_VOC   5000
#define C1      16
#define C2      32
#define KW2     5
#define DN      352
#define VTILES  313   // ceil(5000/16)
#define TVG     40    // ceil(313/8) v-tile groups per (s,tm)

// ---------------- graph preprocessing ----------------

__global__ void k_init(float* deg, float* e2n) {
    int n = blockIdx.x * blockDim.x + threadIdx.x;
    if (n < N_NODES) {
        deg[n] = 1.0f;                      // self-loop
        #pragma unroll
        for (int f = 0; f < EF; ++f) e2n[n * EF + f] = 0.0f;
    }
}

__global__ void k_edge_scatter(const int* __restrict__ ei, const float* __restrict__ ea,
                               float* deg, float* e2n) {
    int e = blockIdx.x * blockDim.x + threadIdx.x;
    if (e >= E_EDGES) return;
    int s = ei[e], d = ei[E_EDGES + e];
    atomicAdd(&deg[d], 1.0f);
    #pragma unroll
    for (int f = 0; f < EF; ++f) {
        float v = ea[e * EF + f];
        atomicAdd(&e2n[s * EF + f], v);
        if (d != s) atomicAdd(&e2n[d * EF + f], v);
    }
}

__global__ void k_dis(const float* deg, float* dis) {
    int n = blockIdx.x * blockDim.x + threadIdx.x;
    if (n < N_NODES) dis[n] = rsqrtf(deg[n]);
}

// Precompute node_emb @ W0[:256] (100x32) and depth_emb @ W0[:256] (20x32)
__global__ void k_embW(const float* __restrict__ node_emb, const float* __restrict__ depth_emb,
                       const float* __restrict__ W0, float* embW, float* depW) {
    int t = blockIdx.x * blockDim.x + threadIdx.x;
    if (t < 100 * HID) {
        int r = t / HID, c = t % HID;
        float acc = 0.f;
        for (int e = 0; e < EMB; ++e) acc += node_emb[r * EMB + e] * W0[e * HID + c];
        embW[t] = acc;
    } else if (t < 120 * HID) {
        int tt = t - 100 * HID;
        int r = tt / HID, c = tt % HID;
        float acc = 0.f;
        for (int e = 0; e < EMB; ++e) acc += depth_emb[r * EMB + e] * W0[e * HID + c];
        depW[tt] = acc;
    }
}

// Fused layer-0 GEMM: hW0[n,c] = embW[type[n],c] + depW[depth[n],c] + e2n[n,:] @ W0[256:264,c]
__global__ void k_hW0(const int* __restrict__ ntype, const int* __restrict__ ndep,
                      const float* __restrict__ e2n, const float* __restrict__ W0,
                      const float* __restrict__ embW, const float* __restrict__ depW,
                      float* hW) {
    int i = blockIdx.x * blockDim.x + threadIdx.x;
    if (i >= N_NODES * HID) return;
    int n = i / HID, c = i % HID;
    float acc = embW[ntype[n] * HID + c] + depW[ndep[n] * HID + c];
    #pragma unroll
    for (int f = 0; f < EF; ++f) acc += e2n[n * EF + f] * W0[(EMB + f) * HID + c];
    hW[i] = acc;
}

// ---------------- GCN aggregate chain ----------------

__global__ void k_agg_init(const float* __restrict__ hW, const float* __restrict__ dis,
                           float* agg, int C) {
    int i = blockIdx.x * blockDim.x + threadIdx.x;
    if (i >= N_NODES * C) return;
    int n = i / C;
    float s = dis[n];
    agg[i] = hW[i] * s * s;                  // self-loop, norm = dis[n]^2
}

__global__ void k_edge_agg(const int* __restrict__ ei, const float* __restrict__ hW,
                           const float* __restrict__ dis, float* agg, int C) {
    int e = blockIdx.x * blockDim.x + threadIdx.x;
    if (e >= E_EDGES) return;
    int s = ei[e], d = ei[E_EDGES + e];
    float w = dis[s] * dis[d];
    for (int c = 0; c < C; ++c)
        atomicAdd(&agg[d * C + c], hW[s * C + c] * w);
}

__global__ void k_tanh_msg(const float* __restrict__ agg, const float* __restrict__ bias,
                           float* msg, int C, int off) {
    int i = blockIdx.x * blockDim.x + threadIdx.x;
    if (i >= N_NODES * C) return;
    int n = i / C, c = i % C;
    msg[(size_t)n * TOT + off + c] = tanhf(agg[i] + bias[c]);
}

// ---------------- f32 WMMA GEMM for GCN layers 1-2: [N,32] @ [32,32] ----------------
// One wave computes a 16x16 output tile. A layout (16x4 f32): lane L holds
// A[L%16][kh + j] in vgpr j, kh = 2*(L/16). B/C/D per ISA §7.12.2.
__global__ void k_gcn_gemm_wmma(const float* __restrict__ msg, int off,
                                const float* __restrict__ W, float* hW) {
    int wave = (blockIdx.x * blockDim.x + threadIdx.x) >> 5;
    int lane = threadIdx.x & 31;
    int tm = wave >> 1;                      // 0..8191 row tiles
    int tn = wave & 1;                       // 0..1 col tiles
    int m  = tm * 16 + (lane & 15);
    int kh = (lane >> 4) << 1;               // 0 or 2
    const float* arow = msg + (size_t)m * TOT + off;
    v8f c = {0.f, 0.f, 0.f, 0.f, 0.f, 0.f, 0.f, 0.f};
    #pragma unroll
    for (int k0 = 0; k0 < HID; k0 += 4) {
        v2f a, b;
        a.x = arow[k0 + kh + 0];
        a.y = arow[k0 + kh + 1];
        b.x = W[(k0 + kh + 0) * HID + tn * 16 + (lane & 15)];
        b.y = W[(k0 + kh + 1) * HID + tn * 16 + (lane & 15)];
        c = __builtin_amdgcn_wmma_f32_16x16x4_f32(false, a, false, b, (short)0, c, false, false);
    }
    int row0 = tm * 16 + ((lane >> 4) << 3);
    int col  = tn * 16 + (lane & 15);
    #pragma unroll
    for (int r = 0; r < 8; ++r)
        hW[(size_t)(row0 + r) * HID + col] = c[r];
}

// Layer 3: [N,32] @ [32,1] gemv
__global__ void k_gemv3(const float* __restrict__ msg, const float* __restrict__ W3, float* hW) {
    int n = blockIdx.x * blockDim.x + threadIdx.x;
    if (n >= N_NODES) return;
    const float* h = msg + (size_t)n * TOT + 64;
    float acc = 0.f;
    #pragma unroll
    for (int c = 0; c < HID; ++c) acc += h[c] * W3[c];
    hW[n] = acc;
}

// ---------------- sort pooling: per-graph bitonic sort of 512 by last channel ----------------
__global__ void k_sortpool(const float* __restrict__ msg, float* pooled) {
    __shared__ float sv[NPG];
    __shared__ int   si[NPG];
    int g = blockIdx.x, tid = threadIdx.x;
    sv[tid] = msg[(size_t)(g * NPG + tid) * TOT + (TOT - 1)];
    si[tid] = tid;
    __syncthreads();
    for (int k = 2; k <= NPG; k <<= 1) {
        for (int j = k >> 1; j > 0; j >>= 1) {
            int ix = tid ^ j;
            if (ix > tid) {
                bool up = ((tid & k) == 0);
                float v0 = sv[tid], v1 = sv[ix];
                int   i0 = si[tid], i1 = si[ix];
                // "before": value desc, tie -> index asc (matches argsort(-v) stability)
                bool before01 = (v0 > v1) || (v0 == v1 && i0 < i1);
                bool doswap = up ? (!before01) : before01;
                if (doswap) { sv[tid] = v1; sv[ix] = v0; si[tid] = i1; si[ix] = i0; }
            }
            __syncthreads();
        }
    }
    for (int t = tid; t < K_POOL * TOT; t += NPG) {
        int k = t / TOT, c = t % TOT;
        pooled[(size_t)g * K_POOL * TOT + t] = msg[(size_t)(g * NPG + si[k]) * TOT + c];
    }
}

// ---------------- CNN head ----------------

__global__ void k_conv1pool(const float* __restrict__ pooled, const float* __restrict__ w,
                            const float* __restrict__ bias, float* p1) {
    int i = blockIdx.x * blockDim.x + threadIdx.x;
    if (i >= B_GR * C1 * 15) return;
    int j = i % 15, c = (i / 15) % C1, b = i / (15 * C1);
    float m = -INFINITY;
    #pragma unroll
    for (int h = 0; h < 2; ++h) {
        int k = j * 2 + h;
        const float* row = pooled + ((size_t)b * K_POOL + k) * TOT;
        float acc = bias[c];
        for (int t = 0; t < TOT; ++t) acc += row[t] * w[c * TOT + t];
        m = fmaxf(m, fmaxf(acc, 0.f));
    }
    p1[i] = m;                               // [b][c1][j], j in 0..14
}

__global__ void k_conv2(const float* __restrict__ p1, const float* __restrict__ w,
                        const float* __restrict__ bias, float* dense) {
    int i = blockIdx.x * blockDim.x + threadIdx.x;
    if (i >= B_GR * C2 * 11) return;
    int j = i % 11, o = (i / 11) % C2, b = i / (11 * C2);
    float acc = bias[o];
    for (int ci = 0; ci < C1; ++ci)
        #pragma unroll
        for (int kw = 0; kw < KW2; ++kw)
            acc += p1[((size_t)b * C1 + ci) * 15 + j + kw] * w[((size_t)o * C1 + ci) * KW2 + kw];
    dense[(size_t)b * DN + o * 11 + j] = fmaxf(acc, 0.f);
}

// ---------------- prediction GEMM: out[s,b,v] = dense[b,:] @ pred_w[s,:,v] + pred_b[s,v] ----
// Block = (s, tm, group of 8 v-tiles). The 16x352 A tile (contiguous: 5632 f32) is staged
// into LDS once per block by the Tensor Data Mover (TDM), synchronized with TENSORcnt,
// then all 8 waves feed their WMMAs from ds_load.
__global__ void k_pred_wmma(const float* __restrict__ dense, const float* __restrict__ pw,
                            const float* __restrict__ pb, float* __restrict__ out) {
    __shared__ float Atile[16 * DN];         // 22528 B; sole LDS object -> offset 0
    int blk = blockIdx.x;
    int s   = blk / (16 * TVG);
    int rem = blk % (16 * TVG);
    int tm  = rem / TVG;
    int tvg = rem % TVG;

    if ((threadIdx.x >> 5) == 0) {           // wave 0 issues the tensor DMA
        unsigned long long ga =
            (unsigned long long)(const void*)(dense + (size_t)tm * 16 * DN);
        const unsigned ELEMS = 16 * DN;      // 5632 contiguous f32 elements
        // D# group 0: count=1, lds_addr=0, global_addr, type=2 (ISA §8.3)
        u32x4 g0 = { 1u, 0u,
                     (unsigned)(ga & 0xFFFFFFFFu),
                     (unsigned)((ga >> 32) & 0x1FFFFFFu) | (2u << 30) };
        // D# group 1: data_size=2 (4B); tensor_dim0 = tile_dim0 = ELEMS; 1-D tile (§8.4)
        i32x8 g1 = { (int)0x00020000,
                     (int)((ELEMS & 0xFFFFu) << 16),   // tensor_dim0[15:0] at bits 63:48
                     (int)((ELEMS >> 16) & 0xFFFFu),   // tensor_dim0[31:16]; tensor_dim1=0
                     (int)((ELEMS & 0xFFFFu) << 16),   // tile_dim0 at bits 127:112
                     0,                                // tile_dim1 = tile_dim2 = 0 (unused)
                     (int)ELEMS,                       // tensor_dim0_stride low 32
                     0, 0 };
        i32x4 gz = { 0, 0, 0, 0 };
#if __clang_major__ >= 23
        i32x8 gz8 = { 0, 0, 0, 0, 0, 0, 0, 0 };
        __builtin_amdgcn_tensor_load_to_lds(g0, g1, gz, gz, gz8, 0);
#else
        __builtin_amdgcn_tensor_load_to_lds(g0, g1, gz, gz, 0);
#endif
        __builtin_amdgcn_s_wait_tensorcnt(0);
    }
    __syncthreads();

    int wave = threadIdx.x >> 5;
    int lane = threadIdx.x & 31;
    int tv = tvg * 8 + wave;
    if (tv >= VTILES) return;                // after the barrier: safe wave-uniform exit

    int v  = tv * 16 + (lane & 15);
    int kh = (lane >> 4) << 1;
    const float* As = Atile + (size_t)(lane & 15) * DN;   // A rows from LDS
    const float* Bp = pw + (size_t)s * DN * V_VOC;
    bool vok = (v < V_VOC);
    size_t bcol = (size_t)(vok ? v : 0);
    v8f c = {0.f, 0.f, 0.f, 0.f, 0.f, 0.f, 0.f, 0.f};
    for (int k0 = 0; k0 < DN; k0 += 4) {
        if (k0 + 8 < DN)
            __builtin_prefetch(&Bp[(size_t)(k0 + 8 + kh) * V_VOC + bcol], 0, 1);
        v2f a, b;
        a.x = As[k0 + kh + 0];
        a.y = As[k0 + kh + 1];
        b.x = vok ? Bp[(size_t)(k0 + kh + 0) * V_VOC + bcol] : 0.f;
        b.y = vok ? Bp[(size_t)(k0 + kh + 1) * V_VOC + bcol] : 0.f;
        c = __builtin_amdgcn_wmma_f32_16x16x4_f32(false, a, false, b, (short)0, c, false, false);
    }
    if (vok) {
        int row0 = tm * 16 + ((lane >> 4) << 3);
        float bias = pb[(size_t)s * V_VOC + v];
        #pragma unroll
        for (int r = 0; r < 8; ++r)
            out[((size_t)s * B_GR + row0 + r) * V_VOC + v] = c[r] + bias;
    }
}

// ---------------- launcher ----------------

extern "C" void kernel_launch(void* const* d_in, const int* in_sizes, int n_in,
                              void* d_out, int out_size, void* d_ws, size_t ws_size,
                              hipStream_t stream) {
    (void)in_sizes; (void)n_in; (void)out_size; (void)ws_size;
    const int*   node_type  = (const int*)d_in[0];
    const int*   node_depth = (const int*)d_in[1];
    const int*   edge_index = (const int*)d_in[2];
    const float* edge_attr  = (const float*)d_in[3];
    // d_in[4] = batch (unused: equal-sized graphs)
    const float* node_emb = (const float*)d_in[5];
    const float* depth_emb= (const float*)d_in[6];
    const float* W0 = (const float*)d_in[7];   const float* b0 = (const float*)d_in[8];
    const float* W1 = (const float*)d_in[9];   const float* b1 = (const float*)d_in[10];
    const float* W2 = (const float*)d_in[11];  const float* b2 = (const float*)d_in[12];
    const float* W3 = (const float*)d_in[13];  const float* b3 = (const float*)d_in[14];
    const float* c1w = (const float*)d_in[15]; const float* c1b = (const float*)d_in[16];
    const float* c2w = (const float*)d_in[17]; const float* c2b = (const float*)d_in[18];
    const float* pw  = (const float*)d_in[19]; const float* pb  = (const float*)d_in[20];
    float* out = (float*)d_out;

    char* ws = (char*)d_ws;
    size_t off = 0;
    auto alloc = [&](size_t bytes) -> float* {
        float* p = (float*)(ws + off);
        off = (off + bytes + 255) & ~(size_t)255;
        return p;
    };
    float* e2n    = alloc((size_t)N_NODES * EF * 4);
    float* deg    = alloc((size_t)N_NODES * 4);
    float* dis    = alloc((size_t)N_NODES * 4);
    float* embW   = alloc(100 * HID * 4);
    float* depW   = alloc(20 * HID * 4);
    float* hW     = alloc((size_t)N_NODES * HID * 4);
    float* agg    = alloc((size_t)N_NODES * HID * 4);
    float* msg    = alloc((size_t)N_NODES * TOT * 4);
    float* pooled = alloc((size_t)B_GR * K_POOL * TOT * 4);
    float* p1     = alloc((size_t)B_GR * C1 * 15 * 4);
    float* dense  = alloc((size_t)B_GR * DN * 4);

    const int T = 256;
    // preprocessing
    k_init<<<(N_NODES + T - 1) / T, T, 0, stream>>>(deg, e2n);
    k_edge_scatter<<<(E_EDGES + T - 1) / T, T, 0, stream>>>(edge_index, edge_attr, deg, e2n);
    k_dis<<<(N_NODES + T - 1) / T, T, 0, stream>>>(deg, dis);
    k_embW<<<(120 * HID + T - 1) / T, T, 0, stream>>>(node_emb, depth_emb, W0, embW, depW);
    k_hW0<<<(N_NODES * HID + T - 1) / T, T, 0, stream>>>(node_type, node_depth, e2n, W0,
                                                         embW, depW, hW);
    // layer 0 aggregate
    k_agg_init<<<(N_NODES * HID + T - 1) / T, T, 0, stream>>>(hW, dis, agg, HID);
    k_edge_agg<<<(E_EDGES + T - 1) / T, T, 0, stream>>>(edge_index, hW, dis, agg, HID);
    k_tanh_msg<<<(N_NODES * HID + T - 1) / T, T, 0, stream>>>(agg, b0, msg, HID, 0);
    // layer 1: WMMA GEMM then aggregate
    k_gcn_gemm_wmma<<<2048, T, 0, stream>>>(msg, 0, W1, hW);
    k_agg_init<<<(N_NODES * HID + T - 1) / T, T, 0, stream>>>(hW, dis, agg, HID);
    k_edge_agg<<<(E_EDGES + T - 1) / T, T, 0, stream>>>(edge_index, hW, dis, agg, HID);
    k_tanh_msg<<<(N_NODES * HID + T - 1) / T, T, 0, stream>>>(agg, b1, msg, HID, 32);
    // layer 2
    k_gcn_gemm_wmma<<<2048, T, 0, stream>>>(msg, 32, W2, hW);
    k_agg_init<<<(N_NODES * HID + T - 1) / T, T, 0, stream>>>(hW, dis, agg, HID);
    k_edge_agg<<<(E_EDGES + T - 1) / T, T, 0, stream>>>(edge_index, hW, dis, agg, HID);
    k_tanh_msg<<<(N_NODES * HID + T - 1) / T, T, 0, stream>>>(agg, b2, msg, HID, 64);
    // layer 3 (output width 1)
    k_gemv3<<<(N_NODES + T - 1) / T, T, 0, stream>>>(msg, W3, hW);
    k_agg_init<<<(N_NODES + T - 1) / T, T, 0, stream>>>(hW, dis, agg, 1);
    k_edge_agg<<<(E_EDGES + T - 1) / T, T, 0, stream>>>(edge_index, hW, dis, agg, 1);
    k_tanh_msg<<<(N_NODES + T - 1) / T, T, 0, stream>>>(agg, b3, msg, 1, 96);
    // sort pooling + CNN head
    k_sortpool<<<B_GR, NPG, 0, stream>>>(msg, pooled);
    k_conv1pool<<<(B_GR * C1 * 15 + T - 1) / T, T, 0, stream>>>(pooled, c1w, c1b, p1);
    k_conv2<<<(B_GR * C2 * 11 + T - 1) / T, T, 0, stream>>>(p1, c2w, c2b, dense);
    // prediction: 5 * 16 * 40 blocks, 8 waves each (one 16x16 tile per wave)
    k_pred_wmma<<<S_SEQ * 16 * TVG, T, 0, stream>>>(dense, pw, pb, out);
}